// ImprovedChunkFusion_40226663694454
// MI455X (gfx1250) — compile-verified
//
#include <hip/hip_runtime.h>
#include <hip/hip_bf16.h>
#include <math.h>

#define N_PTS   8192
#define NCHUNK  8
#define STRIDE  6144
#define TOTAL   (STRIDE * (NCHUNK - 1) + N_PTS)   // 51200
#define FADE    1228                               // int(8192*0.3/2)
#define CTILE   2048                               // columns per block (LDS segment)
#define NSEG    (N_PTS / CTILE)                    // 4 column segments

typedef __attribute__((ext_vector_type(2))) float v2f;
typedef __attribute__((ext_vector_type(8))) float v8f;

// order-preserving float -> u32 key (ascending)
__device__ __forceinline__ unsigned int fkey(float f) {
  unsigned int b = __float_as_uint(f);
  return (b & 0x80000000u) ? ~b : (b | 0x80000000u);
}

// ---------------------------------------------------------------------------
// Kernel 1: fused distance + row-argmin via V_WMMA_F32_16X16X4_F32.
// score(r,c) = ||c||^2 - 2 p_r . c_c  (row-constant ||p||^2 dropped; argmin
// identical to cdist argmin). A-row = [px,py,pz,1], B-col = [-2cx,-2cy,-2cz,cc].
// Column dim split 4-way across blocks for 2048-wave parallelism; segments are
// combined with a packed (key<<32|col) u64 atomicMin — order-independent
// (deterministic) and ties resolve to the lowest column index, matching argmin.
// Inner loop: power-of-two trip count, wrap-around prefetch (branch-free body),
// depth-2 LDS double-buffer so each ds_load_b64 is issued 2 iterations before
// its WMMA consumes it.
// ---------------------------------------------------------------------------
__global__ __launch_bounds__(256) void argmin_wmma_kernel(
    const float* __restrict__ P, const float* __restrict__ C,
    unsigned long long* __restrict__ minbuf) {
  __shared__ float2 e0[CTILE];   // lanes 0-15 : (K0,K2) = (-2x, -2z)
  __shared__ float2 e1[CTILE];   // lanes 16-31: (K1,K3) = (-2y, ||c||^2)

  const int lane  = threadIdx.x & 31;
  const int wave  = threadIdx.x >> 5;
  const int r0    = (blockIdx.y * 8 + wave) * 16;  // 64 row-blocks * 8 waves * 16 = 8192 rows
  const int cbase = blockIdx.x * CTILE;            // 4 column segments
  const int half  = lane >> 4;
  const int l     = lane & 15;

  // A fragment (16x4 f32): VGPR0 = K0(lanes0-15)/K2(lanes16-31), VGPR1 = K1/K3
  v2f a;
  {
    const float* p = P + (size_t)(r0 + l) * 3;
    if (half == 0) { a.x = p[0]; a.y = p[1]; }   // K0 = px, K1 = py
    else           { a.x = p[2]; a.y = 1.0f; }   // K2 = pz, K3 = 1
  }

  // stage this block's column segment into LDS (transformed)
  for (int j = threadIdx.x; j < CTILE; j += 256) {
    const float* c = C + (size_t)(cbase + j) * 3;
    float x = c[0], y = c[1], z = c[2];
    e0[j] = make_float2(-2.0f * x, -2.0f * z);
    e1[j] = make_float2(-2.0f * y, x * x + y * y + z * z);
  }
  __syncthreads();

  const float2* __restrict__ bsrc = (half == 0) ? e0 : e1;

  float minv[8];
  int   mini[8];
#pragma unroll
  for (int k = 0; k < 8; ++k) { minv[k] = 3.4e38f; mini[k] = 0; }

  const int NT = CTILE / 16;                  // 128 tiles (power of two)
  float2 bv0 = bsrc[l];                       // tile 0
  float2 bv1 = bsrc[16 + l];                  // tile 1
#pragma unroll 2
  for (int t = 0; t < NT; ++t) {              // branch-free body, even trip count
    v2f b; b.x = bv0.x; b.y = bv0.y;
    bv0 = bv1;
    bv1 = bsrc[((t + 2) & (NT - 1)) * 16 + l];   // issued 2 iterations ahead (wraps, harmless)

    v8f acc = {};
    v8f d = __builtin_amdgcn_wmma_f32_16x16x4_f32(
        false, a, false, b, (short)0, acc, false, false);

    const int col = cbase + t * 16 + l;
#pragma unroll
    for (int k = 0; k < 8; ++k) {
      float v = d[k];
      if (v < minv[k]) { minv[k] = v; mini[k] = col; }  // ascending scan => lowest idx on tie
    }
  }

  // cross-lane argmin within each 16-lane half, then combine segments via u64 atomicMin
#pragma unroll
  for (int k = 0; k < 8; ++k) {
    float v   = minv[k];
    int   idx = mini[k];
#pragma unroll
    for (int m = 8; m >= 1; m >>= 1) {
      float ov = __shfl_xor(v, m, 32);
      int   oi = __shfl_xor(idx, m, 32);
      if (ov < v || (ov == v && oi < idx)) { v = ov; idx = oi; }
    }
    if (l == 0) {
      unsigned long long packed =
          ((unsigned long long)fkey(v) << 32) | (unsigned int)idx;
      atomicMin(&minbuf[r0 + k + half * 8], packed);
    }
  }
}

// ---------------------------------------------------------------------------
// Kernel 2: fully fused post-processing for one pair:
//   matched gather + weight-predictor MLP + fuse  ->  fusedL (LDS, halo 6)
//   conv1 (K=5,pad2) + relu                       ->  h1L   (LDS, halo 4)
//   conv2 (K=5,pad2) + relu                       ->  h2L   (LDS, halo 2)
//   conv3 (K=5,pad2)                              ->  T (global temp)
// Also: aux[g] = -1 (scatter init) and S_next[g] = C[g] (scatter base).
// ---------------------------------------------------------------------------
#define SB   128                 // output positions per block
#define FH   6                   // fused halo
#define FSZ  (SB + 2 * FH)       // 140
#define H1SZ (SB + 8)            // 136
#define H2SZ (SB + 4)            // 132

__global__ __launch_bounds__(SB) void smoother_kernel(
    const float* __restrict__ P, const float* __restrict__ C,
    const unsigned long long* __restrict__ minbuf,
    const float* __restrict__ pw1, const float* __restrict__ pb1,  // [32,6],[32]
    const float* __restrict__ pw2, const float* __restrict__ pb2,  // [16,32],[16]
    const float* __restrict__ pw3, const float* __restrict__ pb3,  // [16],[1]
    const float* __restrict__ w1, const float* __restrict__ b1,    // [32,3,5],[32]
    const float* __restrict__ w2, const float* __restrict__ b2,    // [32,32,5],[32]
    const float* __restrict__ w3, const float* __restrict__ b3,    // [3,32,5],[3]
    float* __restrict__ T, float* __restrict__ S_next,
    int* __restrict__ aux) {
  __shared__ float fusedL[FSZ * 3];
  __shared__ float h1L[32][H1SZ];
  __shared__ float h2L[32][H2SZ];

  const int tid = threadIdx.x;
  const int B0  = blockIdx.x * SB;

  // ---- stage 1: fuse (with halo), zero outside [0,N) ----
  for (int j = tid; j < FSZ; j += SB) {
    int g = B0 - FH + j;
    float f0 = 0.f, f1 = 0.f, f2 = 0.f;
    if (g >= 0 && g < N_PTS) {
      float px = P[g * 3 + 0], py = P[g * 3 + 1], pz = P[g * 3 + 2];
      int mj = (int)(unsigned int)(minbuf[g] & 0xFFFFFFFFull);
      float mx = C[mj * 3 + 0], my = C[mj * 3 + 1], mz = C[mj * 3 + 2];
      float feat[6] = {px, py, pz, mx, my, mz};
      float ha[32];
#pragma unroll
      for (int co = 0; co < 32; ++co) {
        float s = pb1[co];
#pragma unroll
        for (int ci = 0; ci < 6; ++ci) s += pw1[co * 6 + ci] * feat[ci];
        ha[co] = fmaxf(s, 0.0f);
      }
      float hb[16];
#pragma unroll
      for (int co = 0; co < 16; ++co) {
        float s = pb2[co];
#pragma unroll
        for (int ci = 0; ci < 32; ++ci) s += pw2[co * 32 + ci] * ha[ci];
        hb[co] = fmaxf(s, 0.0f);
      }
      float o = pb3[0];
#pragma unroll
      for (int ci = 0; ci < 16; ++ci) o += pw3[ci] * hb[ci];
      float w = 1.0f / (1.0f + __expf(-o));
      f0 = w * px + (1.0f - w) * mx;
      f1 = w * py + (1.0f - w) * my;
      f2 = w * pz + (1.0f - w) * mz;
    }
    fusedL[j * 3 + 0] = f0;
    fusedL[j * 3 + 1] = f1;
    fusedL[j * 3 + 2] = f2;
  }
  __syncthreads();

  // ---- stage 2: conv1 + relu -> h1L (halo 4) ----
  for (int j = tid; j < H1SZ; j += SB) {
    int g = B0 - 4 + j;
    float acc[32];
    bool inr = (g >= 0 && g < N_PTS);
#pragma unroll
    for (int co = 0; co < 32; ++co) acc[co] = inr ? b1[co] : 0.0f;
    if (inr) {
#pragma unroll
      for (int dk = 0; dk < 5; ++dk) {
        int jf = j + dk;           // (g + dk - 2) - (B0 - 6)
        float x0 = fusedL[jf * 3 + 0];
        float x1 = fusedL[jf * 3 + 1];
        float x2 = fusedL[jf * 3 + 2];
#pragma unroll
        for (int co = 0; co < 32; ++co) {
          acc[co] += w1[co * 15 + 0 * 5 + dk] * x0
                   + w1[co * 15 + 1 * 5 + dk] * x1
                   + w1[co * 15 + 2 * 5 + dk] * x2;
        }
      }
    }
#pragma unroll
    for (int co = 0; co < 32; ++co)
      h1L[co][j] = inr ? fmaxf(acc[co], 0.0f) : 0.0f;
  }
  __syncthreads();

  // ---- stage 3: conv2 + relu -> h2L (halo 2) ----
  for (int j = tid; j < H2SZ; j += SB) {
    int g = B0 - 2 + j;
    float acc[32];
    bool inr = (g >= 0 && g < N_PTS);
#pragma unroll
    for (int co = 0; co < 32; ++co) acc[co] = inr ? b2[co] : 0.0f;
    if (inr) {
      for (int ci = 0; ci < 32; ++ci) {
#pragma unroll
        for (int dk = 0; dk < 5; ++dk) {
          float xv = h1L[ci][j + dk];   // (g + dk - 2) - (B0 - 4)
#pragma unroll
          for (int co = 0; co < 32; ++co)
            acc[co] += w2[co * 160 + ci * 5 + dk] * xv;
        }
      }
    }
#pragma unroll
    for (int co = 0; co < 32; ++co)
      h2L[co][j] = inr ? fmaxf(acc[co], 0.0f) : 0.0f;
  }
  __syncthreads();

  // ---- stage 4: conv3 -> T, plus aux init and S_next = C copy ----
  {
    int g = B0 + tid;
    float a0 = b3[0], a1 = b3[1], a2 = b3[2];
    for (int ci = 0; ci < 32; ++ci) {
#pragma unroll
      for (int dk = 0; dk < 5; ++dk) {
        float xv = h2L[ci][tid + dk];   // (g + dk - 2) - (B0 - 2)
        a0 += w3[0 * 160 + ci * 5 + dk] * xv;
        a1 += w3[1 * 160 + ci * 5 + dk] * xv;
        a2 += w3[2 * 160 + ci * 5 + dk] * xv;
      }
    }
    T[g * 3 + 0] = a0;
    T[g * 3 + 1] = a1;
    T[g * 3 + 2] = a2;
    aux[g] = -1;
    S_next[g * 3 + 0] = C[g * 3 + 0];
    S_next[g * 3 + 1] = C[g * 3 + 1];
    S_next[g * 3 + 2] = C[g * 3 + 2];
  }
}

// ---------------------------------------------------------------------------
// Deterministic scatter c.at[mi].set(sm): winner per destination = max source
// row (order-independent atomicMax); finalize commits both outputs and resets
// minbuf for the next pair's argmin.
// ---------------------------------------------------------------------------
__global__ __launch_bounds__(256) void aux_scatter_kernel(
    const unsigned long long* __restrict__ minbuf, int* __restrict__ aux) {
  int r = blockIdx.x * blockDim.x + threadIdx.x;
  if (r < N_PTS) {
    int dest = (int)(unsigned int)(minbuf[r] & 0xFFFFFFFFull);
    atomicMax(&aux[dest], r);
  }
}

__global__ __launch_bounds__(256) void finalize_kernel(
    const float* __restrict__ T, const int* __restrict__ aux,
    float* __restrict__ S_prev, float* __restrict__ S_next,
    unsigned long long* __restrict__ minbuf) {
  int j = blockIdx.x * blockDim.x + threadIdx.x;
  if (j >= N_PTS) return;
  S_prev[j * 3 + 0] = T[j * 3 + 0];
  S_prev[j * 3 + 1] = T[j * 3 + 1];
  S_prev[j * 3 + 2] = T[j * 3 + 2];
  int r = aux[j];
  if (r >= 0) {
    S_next[j * 3 + 0] = T[r * 3 + 0];
    S_next[j * 3 + 1] = T[r * 3 + 1];
    S_next[j * 3 + 2] = T[r * 3 + 2];
  }
  minbuf[j] = ~0ull;   // reset for next pair's argmin
}

__global__ __launch_bounds__(256) void minbuf_init_kernel(
    unsigned long long* __restrict__ minbuf) {
  int j = blockIdx.x * blockDim.x + threadIdx.x;
  if (j < N_PTS) minbuf[j] = ~0ull;
}

// ---------------------------------------------------------------------------
// Final fade-weighted merge; each output position pulls from <=2 chunks.
// ---------------------------------------------------------------------------
__device__ __forceinline__ float fade_w(int n) {
  if (n < FADE)          return 0.1f + 0.9f * (float)n / (float)(FADE - 1);
  if (n >= N_PTS - FADE) return 1.0f - 0.9f * (float)(n - (N_PTS - FADE)) / (float)(FADE - 1);
  return 1.0f;
}

__global__ __launch_bounds__(256) void merge_kernel(
    const float* __restrict__ S /*[8,N,3]*/, float* __restrict__ out) {
  int t = blockIdx.x * blockDim.x + threadIdx.x;
  if (t >= TOTAL) return;
  float a0 = 0.f, a1 = 0.f, a2 = 0.f, ws = 0.f;
#pragma unroll
  for (int c = 0; c < NCHUNK; ++c) {
    int n = t - c * STRIDE;
    if (n >= 0 && n < N_PTS) {
      float w = fade_w(n);
      const float* p = S + ((size_t)c * N_PTS + n) * 3;
      a0 += w * p[0]; a1 += w * p[1]; a2 += w * p[2];
      ws += w;
    }
  }
  float inv = 1.0f / fmaxf(ws, 1e-8f);
  out[t * 3 + 0] = a0 * inv;
  out[t * 3 + 1] = a1 * inv;
  out[t * 3 + 2] = a2 * inv;
}

// ---------------------------------------------------------------------------
extern "C" void kernel_launch(void* const* d_in, const int* in_sizes, int n_in,
                              void* d_out, int out_size, void* d_ws, size_t ws_size,
                              hipStream_t stream) {
  const float* chunks = (const float*)d_in[0];   // [8,8192,3]
  const float* bs_w1  = (const float*)d_in[1];
  const float* bs_b1  = (const float*)d_in[2];
  const float* bs_w2  = (const float*)d_in[3];
  const float* bs_b2  = (const float*)d_in[4];
  const float* bs_w3  = (const float*)d_in[5];
  const float* bs_b3  = (const float*)d_in[6];
  const float* wp_w1  = (const float*)d_in[7];
  const float* wp_b1  = (const float*)d_in[8];
  const float* wp_w2  = (const float*)d_in[9];
  const float* wp_b2  = (const float*)d_in[10];
  const float* wp_w3  = (const float*)d_in[11];
  const float* wp_b3  = (const float*)d_in[12];

  // workspace layout
  float*              S      = (float*)d_ws;                        // 8*N*3 floats
  float*              T      = S + (size_t)NCHUNK * N_PTS * 3;      // N*3 floats
  unsigned long long* minbuf = (unsigned long long*)(T + (size_t)N_PTS * 3); // N u64
  int*                aux    = (int*)(minbuf + N_PTS);              // N ints

  // smoothed[0] = chunks[0]; minbuf -> +inf keys
  hipMemcpyAsync(S, chunks, (size_t)N_PTS * 3 * sizeof(float),
                 hipMemcpyDeviceToDevice, stream);
  minbuf_init_kernel<<<32, 256, 0, stream>>>(minbuf);

  for (int i = 1; i < NCHUNK; ++i) {
    const float* Ci    = chunks + (size_t)i * N_PTS * 3;
    float*       Sprev = S + (size_t)(i - 1) * N_PTS * 3;
    float*       Snext = S + (size_t)i * N_PTS * 3;

    dim3 agrid(NSEG, N_PTS / 128);   // 4 column segments x 64 row blocks
    argmin_wmma_kernel<<<agrid, 256, 0, stream>>>(Sprev, Ci, minbuf);
    smoother_kernel<<<N_PTS / SB, SB, 0, stream>>>(
        Sprev, Ci, minbuf,
        wp_w1, wp_b1, wp_w2, wp_b2, wp_w3, wp_b3,
        bs_w1, bs_b1, bs_w2, bs_b2, bs_w3, bs_b3,
        T, Snext, aux);
    aux_scatter_kernel<<<32, 256, 0, stream>>>(minbuf, aux);
    finalize_kernel<<<32, 256, 0, stream>>>(T, aux, Sprev, Snext, minbuf);
  }

  merge_kernel<<<(TOTAL + 255) / 256, 256, 0, stream>>>(S, (float*)d_out);
}